// WHVIStackedMatrix_45861660786918
// MI455X (gfx1250) — compile-verified
//
#include <hip/hip_runtime.h>
#include <hip/hip_bf16.h>
#include <math.h>

typedef __attribute__((ext_vector_type(2))) float v2f;
typedef __attribute__((ext_vector_type(8))) float v8f;
typedef __attribute__((ext_vector_type(4))) unsigned int v4u;
typedef __attribute__((ext_vector_type(8))) int v8i;
typedef __attribute__((ext_vector_type(4))) int v4i;

#define D 1024
#define STACKN 4
#define BATCHN 8192
#define TROWS 16      // batch rows per workgroup tile
#define LSTR 1092     // LDS row stride in floats: 1024 + 4*17 (TDM pad), %8 != 0
#define NW 4          // waves per workgroup
#define NT 128        // threads per workgroup

// Compute-tile swizzle: keeps 16-runs contiguous (c-axis), spreads the
// stride-16 m-axis walks across banks.
__device__ __forceinline__ int sw(int j) { return j + ((j >> 6) << 2); }
// TDM-load-tile swizzle: matches TDM pad_interval=256dw, pad_amount=17dw.
__device__ __forceinline__ int sw2(int j) { return j + (j >> 8) * 17; }

__device__ __forceinline__ float hsign(int i, int j) {
  return (__builtin_popcount(i & j) & 1) ? -1.0f : 1.0f;
}

// D = A(16x4) * B(4x16) + C using fp32 WMMA
__device__ __forceinline__ v8f wmma4(v2f a, v2f b, v8f c) {
  return __builtin_amdgcn_wmma_f32_16x16x4_f32(
      /*neg_a=*/false, a, /*neg_b=*/false, b,
      /*c_mod=*/(short)0, c, /*reuse_a=*/false, /*reuse_b=*/false);
}

// H16 applied on the low 4 feature bits (c): 16x16 blocks contiguous in LDS.
__device__ __forceinline__ void h16_lo(float* ts, int wave, int lane, const v2f* Bf) {
  const int r  = lane & 15;          // A: M (batch row within tile)
  const int kh = (lane >> 4) << 1;   // A: K half (0 or 2)
  const int n  = lane & 15;          // D: N
  const int mo = (lane >> 4) << 3;   // D: row offset (0 or 8)
  // per-iteration swizzled offset advances by exactly 68 floats (64 + 4)
  const float* src = ts + r * LSTR + sw(wave << 4) + kh;
  float*       dst = ts + sw((wave << 4) + n) + mo * LSTR;
  for (int t = 0; t < 64 / NW; ++t) {
    v8f acc = {};
#pragma unroll
    for (int k = 0; k < 4; ++k) {
      const float* p = src + 4 * k;
      v2f av; av.x = p[0]; av.y = p[1];
      acc = wmma4(av, Bf[k], acc);
    }
#pragma unroll
    for (int i = 0; i < 8; ++i)
      dst[i * LSTR] = acc[i];
    src += NW * 16 + 4;
    dst += NW * 16 + 4;
  }
}

// H16 applied on the middle 4 feature bits (m): block columns stride-16.
__device__ __forceinline__ void h16_mid(float* ts, int wave, int lane, const v2f* Bf) {
  const int r  = lane & 15;
  const int kh = (lane >> 4) << 1;
  const int n  = lane & 15;
  const int mo = (lane >> 4) << 3;
  for (int blk = wave; blk < 64; blk += NW) {
    const int u  = blk >> 4;          // 0..3
    const int c  = blk & 15;          // 0..15
    const int jb = (u << 8) + c;      // feature = jb + 16*m
    const float* srcRow = ts + r * LSTR;
    v8f acc = {};
#pragma unroll
    for (int k = 0; k < 4; ++k) {
      const int m0 = 4 * k + kh;
      v2f av;
      av.x = srcRow[sw(jb + (m0 << 4))];
      av.y = srcRow[sw(jb + ((m0 + 1) << 4))];
      acc = wmma4(av, Bf[k], acc);
    }
    float* dst = ts + sw(jb + (n << 4)) + mo * LSTR;
#pragma unroll
    for (int i = 0; i < 8; ++i)
      dst[i * LSTR] = acc[i];
  }
}

__global__ void __launch_bounds__(NT)
whvi_fwht_wmma(const float* __restrict__ x,  const float* __restrict__ s1,
               const float* __restrict__ s2, const float* __restrict__ g_mu,
               const float* __restrict__ g_rho, const float* __restrict__ eps,
               float* __restrict__ out)
{
  extern __shared__ float lds[];
  float* xs = lds;                   // TROWS x D (TDM sw2 layout), at LDS offset 0
  float* ts = lds + TROWS * LSTR;    // TROWS x D (sw layout): working tile

  const int  tid  = threadIdx.x;
  const int  lane = tid & 31;
  const int  wave = __builtin_amdgcn_readfirstlane(tid >> 5);
  const long rowBase = (long)blockIdx.x * TROWS;

  // Constant Hadamard B fragments: B[k][n] = (-1)^popcount(k&n).
  v2f Bf[4];
  {
    const int n  = lane & 15;
    const int kh = (lane >> 4) << 1;
#pragma unroll
    for (int k = 0; k < 4; ++k) {
      Bf[k].x = hsign(4 * k + kh,     n);
      Bf[k].y = hsign(4 * k + kh + 1, n);
    }
  }

  // ---- TDM: async DMA the 16x1024 f32 x-tile global -> LDS (wave 0 only) ----
  // pad_interval=256 DWORDs, pad_amount=17 DWORDs  =>  LDS element offset
  // r*1092 + j + 17*(j>>8), i.e. row stride LSTR with the sw2 swizzle.
  if (wave == 0) {
    const unsigned long long ga =
        (unsigned long long)(uintptr_t)x + (unsigned long long)rowBase * (D * 4);
    v4u g0;
    g0.x = 1u;                                   // count=1, user mode
    g0.y = 0u;                                   // lds_addr = 0 (xs at LDS base)
    g0.z = (unsigned)(ga & 0xFFFFFFFFu);         // global_addr[31:0]
    g0.w = (unsigned)((ga >> 32) & 0x1FFFFFFu)   // global_addr[56:32]
         | (2u << 30);                           // type = 2 ("image")
    v8i g1;
    g1[0] = (2 << 16)                            // data_size = 4B
          | (1 << 20)                            // pad_enable
          | (7 << 22)                            // pad_interval: 256 DWORDs
          | (16 << 25);                          // pad_amount: 17 DWORDs
    g1[1] = (int)(1024u << 16);                  // tensor_dim0[15:0] = 1024
    g1[2] = (int)(8192u << 16);                  // dim0 hi=0 | tensor_dim1[15:0]=8192
    g1[3] = (int)(1024u << 16);                  // dim1 hi=0 | tile_dim0 = 1024
    g1[4] = TROWS;                               // tile_dim1 = 16, tile_dim2 = 0
    g1[5] = 1024;                                // tensor_dim0_stride[31:0]
    g1[6] = 0;                                   // stride hi / dim1_stride lo
    g1[7] = 0;
    v4i zg4 = {0, 0, 0, 0};                      // groups 2/3 unused (2-D tensor)
    v8i zg8 = {0, 0, 0, 0, 0, 0, 0, 0};          // extra group (6-arg form), unused
    __builtin_amdgcn_tensor_load_to_lds(g0, g1, zg4, zg4, zg8, 0);
    __builtin_amdgcn_s_wait_tensorcnt(0);        // data landed in LDS
  }
  __syncthreads();

  for (int s = 0; s < STACKN; ++s) {
    const float* s1v = s1    + s * D;
    const float* s2v = s2    + s * D;
    const float* gmv = g_mu  + s * D;
    const float* grv = g_rho + s * D;
    const float* epv = eps   + s * D;

    // ts = x-tile * s2  (per-feature); xs uses TDM layout, ts the compute layout
    for (int i = tid; i < TROWS * D; i += NT) {
      const int r = i >> 10;
      const int j = i & 1023;
      ts[r * LSTR + sw(j)] = xs[r * LSTR + sw2(j)] * s2v[j];
    }
    __syncthreads();

    // ---- FWHT #1: H16(c) . H16(m) . H4(u), H4 fused with * g_tilde ----
    h16_lo (ts, wave, lane, Bf);  __syncthreads();
    h16_mid(ts, wave, lane, Bf);  __syncthreads();
    for (int i = tid; i < TROWS * 256; i += NT) {
      const int r  = i >> 8;
      const int j0 = i & 255;
      float* row = ts + r * LSTR;
      const int o0 = sw(j0), o1 = sw(j0 + 256), o2 = sw(j0 + 512), o3 = sw(j0 + 768);
      const float t0 = row[o0], t1 = row[o1], t2 = row[o2], t3 = row[o3];
      const float a0 = t0 + t1, a1 = t0 - t1, a2 = t2 + t3, a3 = t2 - t3;
      const float g0 = gmv[j0      ] + log1pf(__expf(grv[j0      ])) * epv[j0      ];
      const float g1 = gmv[j0 + 256] + log1pf(__expf(grv[j0 + 256])) * epv[j0 + 256];
      const float g2 = gmv[j0 + 512] + log1pf(__expf(grv[j0 + 512])) * epv[j0 + 512];
      const float g3 = gmv[j0 + 768] + log1pf(__expf(grv[j0 + 768])) * epv[j0 + 768];
      row[o0] = (a0 + a2) * g0;
      row[o1] = (a1 + a3) * g1;
      row[o2] = (a0 - a2) * g2;
      row[o3] = (a1 - a3) * g3;
    }
    __syncthreads();

    // ---- FWHT #2: H4 fused with * s1 and direct global store ----
    h16_lo (ts, wave, lane, Bf);  __syncthreads();
    h16_mid(ts, wave, lane, Bf);  __syncthreads();
    for (int i = tid; i < TROWS * 256; i += NT) {
      const int r  = i >> 8;
      const int j0 = i & 255;
      const float* row = ts + r * LSTR;
      const float t0 = row[sw(j0)],       t1 = row[sw(j0 + 256)];
      const float t2 = row[sw(j0 + 512)], t3 = row[sw(j0 + 768)];
      const float a0 = t0 + t1, a1 = t0 - t1, a2 = t2 + t3, a3 = t2 - t3;
      float* op = out + (rowBase + r) * (long)(STACKN * D) + s * D + j0;
      op[0]   = (a0 + a2) * s1v[j0      ];
      op[256] = (a1 + a3) * s1v[j0 + 256];
      op[512] = (a0 - a2) * s1v[j0 + 512];
      op[768] = (a1 - a3) * s1v[j0 + 768];
    }
    __syncthreads();
  }
}

extern "C" void kernel_launch(void* const* d_in, const int* in_sizes, int n_in,
                              void* d_out, int out_size, void* d_ws, size_t ws_size,
                              hipStream_t stream) {
  (void)in_sizes; (void)n_in; (void)out_size; (void)d_ws; (void)ws_size;
  const float* x     = (const float*)d_in[0];
  const float* s1    = (const float*)d_in[1];
  const float* s2    = (const float*)d_in[2];
  const float* g_mu  = (const float*)d_in[3];
  const float* g_rho = (const float*)d_in[4];
  const float* eps   = (const float*)d_in[5];
  float* out = (float*)d_out;

  const size_t shmem = (size_t)2 * TROWS * LSTR * sizeof(float); // ~136.5 KB < 320 KB WGP LDS
  (void)hipFuncSetAttribute((const void*)whvi_fwht_wmma,
                            hipFuncAttributeMaxDynamicSharedMemorySize, (int)shmem);

  dim3 grid(BATCHN / TROWS);   // 512 tiles
  dim3 block(NT);              // 4 waves
  whvi_fwht_wmma<<<grid, block, shmem, stream>>>(x, s1, s2, g_mu, g_rho, eps, out);
}